// AttentionBlock_90237262889673
// MI455X (gfx1250) — compile-verified
//
#include <hip/hip_runtime.h>
#include <hip/hip_bf16.h>

#define NN   100000
#define KK   27
#define CC0  64
#define CC1  32
#define CC2  16
#define LEAK 0.33f
#define EPSV 1e-4f

typedef __attribute__((ext_vector_type(2))) float v2f;
typedef __attribute__((ext_vector_type(8))) float v8f;

// ---------------- helper kernels ----------------

__global__ void pad_copy(const float* __restrict__ f, float* __restrict__ padF) {
    int i = blockIdx.x * 256 + threadIdx.x;
    if (i < (NN + 1) * CC0) padF[i] = (i < NN * CC0) ? f[i] : 0.0f;
}

__global__ void init_small(float* __restrict__ raw1, float* __restrict__ raw2,
                           float* __restrict__ stats1, float* __restrict__ stats2) {
    int t = threadIdx.x;
    if (t < CC1)     raw1[NN * CC1 + t] = 0.0f;   // sentinel row of layer-1 output
    if (t < CC2)     raw2[NN * CC2 + t] = 0.0f;   // sentinel row of layer-2 output
    if (t < 2 * CC1) stats1[t] = 0.0f;
    if (t < 2 * CC2) stats2[t] = 0.0f;
}

// ---------------- WMMA submanifold conv + fused BN-stat reduction ----------------
// D = sum_k gather_A(nbr[:,k]) @ W[k]   using V_WMMA_F32_16X16X4_F32.
// One wave owns a 16-row output tile; 8 waves/block.
// W[k] staged into LDS with double-buffered GLOBAL_LOAD_ASYNC_TO_LDS_B128
// (ASYNCcnt), overlapped with the WMMA K-loop of the previous offset.
// A fragment layout (16x4 f32): lane L holds row M=L&15, K-pair {2*(L>>4), 2*(L>>4)+1}
//  -> one global_load_b64 per lane per K-step, straight from the gathered row.
// C/D layout: lane L reg r -> (M = r + 8*(L>>4), N = L&15).

template <int CIN, int COUT>
__global__ __launch_bounds__(256) void subconv_wmma(
    const float* __restrict__ X,     // (NN+1) x CIN, row NN == 0
    const float* __restrict__ W,     // KK x CIN x COUT
    const int*   __restrict__ nbr,   // NN x KK   (value NN == inactive sentinel)
    float*       __restrict__ Y,     // (NN+1) x COUT, rows < NN written
    float*       __restrict__ stats) // [2*COUT]: sum, then sumsq
{
    constexpr int NT      = COUT / 16;
    constexpr int WFLOATS = CIN * COUT;
    __shared__ __align__(16) float sW[2 * WFLOATS];   // double-buffered W[k]
    __shared__ float sSum[COUT];
    __shared__ float sSqs[COUT];

    const int wave = threadIdx.x >> 5;
    const int lane = threadIdx.x & 31;
    const int half = lane >> 4;        // which K-pair of the 16x4 A fragment
    const int n    = lane & 15;        // output column within N-tile / A row
    const int row0 = blockIdx.x * 128 + wave * 16;
    const int arow_idx = (row0 + n < NN) ? (row0 + n) : (NN - 1); // clamp tail tiles

    const uint32_t lds0 = (uint32_t)(uintptr_t)&sW[0];  // low 32 bits == LDS offset

    // async-stage W[k] into LDS buffer `buf` (16B per issued op, per thread)
    auto stage = [&](int k, int buf) {
        const float*   src = W + (size_t)k * WFLOATS;
        const uint32_t dst = lds0 + (uint32_t)buf * (WFLOATS * 4);
        for (int i = threadIdx.x; i < WFLOATS / 4; i += 256) {
            asm volatile("global_load_async_to_lds_b128 %0, %1, off"
                         :: "v"(dst + (uint32_t)i * 16u), "v"(src + i * 4)
                         : "memory");
        }
    };

    if (threadIdx.x < COUT) { sSum[threadIdx.x] = 0.0f; sSqs[threadIdx.x] = 0.0f; }

    // prologue: stage k=0, fetch first rulebook entry
    stage(0, 0);
    int nb = nbr[arow_idx * KK];
    asm volatile("s_wait_asynccnt 0" ::: "memory");
    __syncthreads();

    v8f acc[NT] = {};

    for (int k = 0; k < KK; ++k) {
        const int cur = k & 1;
        int nb_next = 0;
        if (k + 1 < KK) {
            stage(k + 1, cur ^ 1);                       // overlap with WMMA below
            nb_next = nbr[arow_idx * KK + k + 1];
            __builtin_prefetch(X + (size_t)nb_next * CIN, 0, 3);  // next gathered row
        }

        const float* arow = X + (size_t)nb * CIN;
        const float* wbuf = sW + cur * WFLOATS;

#pragma unroll
        for (int c0 = 0; c0 < CIN; c0 += 4) {
            const int cb = c0 + 2 * half;               // this lane's K-pair base
            v2f a = *(const v2f*)(arow + cb);           // gathered A fragment
#pragma unroll
            for (int t = 0; t < NT; ++t) {
                v2f b;                                  // B fragment: rows cb,cb+1 col t*16+n
                b.x = wbuf[cb * COUT + t * 16 + n];
                b.y = wbuf[(cb + 1) * COUT + t * 16 + n];
                acc[t] = __builtin_amdgcn_wmma_f32_16x16x4_f32(
                    false, a, false, b, (short)0, acc[t], false, false);
            }
        }

        nb = nb_next;
        asm volatile("s_wait_asynccnt 0" ::: "memory"); // next W[k] resident
        __syncthreads();                                // all waves done with `cur`
    }

    // writeback + per-column sum/sumsq for BatchNorm (exclude clamped tail rows)
#pragma unroll
    for (int t = 0; t < NT; ++t) {
        float s = 0.0f, q = 0.0f;
#pragma unroll
        for (int r = 0; r < 8; ++r) {
            int m = row0 + r + 8 * half;
            if (m < NN) {
                float v = acc[t][r];
                Y[(size_t)m * COUT + t * 16 + n] = v;
                s += v;
                q += v * v;
            }
        }
        atomicAdd(&sSum[t * 16 + n], s);   // ds_add_f32
        atomicAdd(&sSqs[t * 16 + n], q);
    }
    __syncthreads();
    if (threadIdx.x < COUT) {
        atomicAdd(&stats[threadIdx.x],        sSum[threadIdx.x]);
        atomicAdd(&stats[COUT + threadIdx.x], sSqs[threadIdx.x]);
    }
}

// ---------------- BN finalize + in-place activation ----------------

__global__ void bn_finalize(const float* __restrict__ stats,
                            const float* __restrict__ g, const float* __restrict__ b,
                            float* __restrict__ params, int C) {
    int c = threadIdx.x;
    if (c < C) {
        float mean  = stats[c] / (float)NN;
        float var   = stats[C + c] / (float)NN - mean * mean;  // biased var, as jnp.var
        float scale = g[c] * rsqrtf(var + EPSV);
        params[c]     = scale;
        params[C + c] = b[c] - mean * scale;
    }
}

__global__ void activate_inplace(float* __restrict__ X, const float* __restrict__ params,
                                 int Cmask, int C, int total) {
    int i = blockIdx.x * 256 + threadIdx.x;
    if (i < total) {
        int c = i & Cmask;
        float y = X[i] * params[c] + params[C + c];
        X[i] = (y >= 0.0f) ? y : LEAK * y;
    }
}

// ---------------- head: Cout=2 conv + softmax>0.5 == argmax ----------------

__global__ void head_mask(const float* __restrict__ H,   // (NN+1) x 16, row NN == 0
                          const float* __restrict__ W3,  // KK x 16 x 2
                          const int*   __restrict__ nbr,
                          float*       __restrict__ mask) {
    int nidx = blockIdx.x * 256 + threadIdx.x;
    if (nidx >= NN) return;
    float o0 = 0.0f, o1 = 0.0f;
    for (int k = 0; k < KK; ++k) {
        int nb = nbr[nidx * KK + k];
        const float* h = H + (size_t)nb * CC2;
#pragma unroll
        for (int c = 0; c < CC2; ++c) {
            float hv = h[c];
            o0 = fmaf(hv, W3[(k * CC2 + c) * 2 + 0], o0);
            o1 = fmaf(hv, W3[(k * CC2 + c) * 2 + 1], o1);
        }
    }
    // softmax(p1) > 0.5  <=>  raw1 > raw0
    mask[nidx] = (o1 > o0) ? 1.0f : 0.0f;
}

__global__ void unpool(const float* __restrict__ mask, const int* __restrict__ parent,
                       float* __restrict__ out, int M) {
    int m = blockIdx.x * 256 + threadIdx.x;
    if (m < M) out[m] = mask[parent[m]];
}

// ---------------- launch ----------------

extern "C" void kernel_launch(void* const* d_in, const int* in_sizes, int n_in,
                              void* d_out, int out_size, void* d_ws, size_t ws_size,
                              hipStream_t stream) {
    const float* features = (const float*)d_in[0];
    const float* W1       = (const float*)d_in[1];
    const float* g1       = (const float*)d_in[2];
    const float* b1       = (const float*)d_in[3];
    const float* W2       = (const float*)d_in[4];
    const float* g2       = (const float*)d_in[5];
    const float* b2       = (const float*)d_in[6];
    const float* W3       = (const float*)d_in[7];
    const int*   nbr      = (const int*)d_in[8];
    const int*   parent   = (const int*)d_in[9];
    const int    M        = in_sizes[9];
    float* out = (float*)d_out;

    float* ws      = (float*)d_ws;
    float* padF    = ws;                                    // (NN+1)*64
    float* raw1    = padF    + (size_t)(NN + 1) * CC0;      // (NN+1)*32
    float* raw2    = raw1    + (size_t)(NN + 1) * CC1;      // (NN+1)*16
    float* maskb   = raw2    + (size_t)(NN + 1) * CC2;      // NN
    float* stats1  = maskb   + NN;                          // 64
    float* params1 = stats1  + 2 * CC1;                     // 64
    float* stats2  = params1 + 2 * CC1;                     // 32
    float* params2 = stats2  + 2 * CC2;                     // 32

    const int convBlocks = (NN + 127) / 128;

    pad_copy<<<((NN + 1) * CC0 + 255) / 256, 256, 0, stream>>>(features, padF);
    init_small<<<1, 128, 0, stream>>>(raw1, raw2, stats1, stats2);

    subconv_wmma<CC0, CC1><<<convBlocks, 256, 0, stream>>>(padF, W1, nbr, raw1, stats1);
    bn_finalize<<<1, CC1, 0, stream>>>(stats1, g1, b1, params1, CC1);
    activate_inplace<<<(NN * CC1 + 255) / 256, 256, 0, stream>>>(raw1, params1, CC1 - 1, CC1, NN * CC1);

    subconv_wmma<CC1, CC2><<<convBlocks, 256, 0, stream>>>(raw1, W2, nbr, raw2, stats2);
    bn_finalize<<<1, CC2, 0, stream>>>(stats2, g2, b2, params2, CC2);
    activate_inplace<<<(NN * CC2 + 255) / 256, 256, 0, stream>>>(raw2, params2, CC2 - 1, CC2, NN * CC2);

    head_mask<<<(NN + 255) / 256, 256, 0, stream>>>(raw2, W3, nbr, maskb);
    unpool<<<(M + 255) / 256, 256, 0, stream>>>(maskb, parent, out, M);
}